// ChamferDistance_81131932221877
// MI455X (gfx1250) — compile-verified
//
#include <hip/hip_runtime.h>
#include <float.h>

typedef __attribute__((ext_vector_type(2))) float v2f;
typedef __attribute__((ext_vector_type(8))) float v8f;

#define WAVES_PER_WG 4
#define WG_THREADS   (WAVES_PER_WG * 32)
#define CHUNK        2048           // y points staged per round: 2048 * 16B = 32 KB LDS
#define TILES_PER_CHUNK (CHUNK / 16)

// One direction of chamfer NN: for each x-row i, min_j ||x_i - y_j||^2 (+argmin).
// WMMA computes the row-shifted distance  t_j = |y_j|^2 - 2 x.y_j  via
//   A row = [x0, x1, x2, 1]          (16x4 fp32)
//   B col = [-2y0, -2y1, -2y2, |y|^2] (4x16 fp32)
//   C     = inline 0
// argmin_j t_j == argmin_j ||x-y_j||^2 (constant |x|^2 shift per row); the
// |x|^2 term and the max(.,0) clamp are applied once at the final write.
__global__ __launch_bounds__(WG_THREADS)
void chamfer_nn_dir_kernel(const float* __restrict__ X, const float* __restrict__ Y,
                           float* __restrict__ dist, int* __restrict__ idx,
                           int Nx, int Ny)
{
    __shared__ float4 sY[CHUNK];

    const int b     = blockIdx.y;
    const int tid   = threadIdx.x;
    const int wave  = tid >> 5;
    const int lane  = tid & 31;
    const int lhalf = lane >> 4;    // 0: lanes 0-15, 1: lanes 16-31
    const int l16   = lane & 15;

    const int r0 = (blockIdx.x * WAVES_PER_WG + wave) * 16;   // 16 rows per wave
    const float* Xb = X + (size_t)b * Nx * 3;
    const float* Yb = Y + (size_t)b * Ny * 3;

    // ---- A matrix (16x4 fp32), ISA layout:
    //   lanes 0-15:  vgpr0 = K0 (x0), vgpr1 = K1 (x1)
    //   lanes 16-31: vgpr0 = K2 (x2), vgpr1 = K3 (1.0)
    const int i0 = r0 + l16;
    const float x0 = Xb[(size_t)i0 * 3 + 0];
    const float x1 = Xb[(size_t)i0 * 3 + 1];
    const float x2 = Xb[(size_t)i0 * 3 + 2];
    const float xn = x0 * x0 + x1 * x1 + x2 * x2;
    v2f a;
    a.x = lhalf ? x2   : x0;
    a.y = lhalf ? 1.0f : x1;

    // Running min/argmin per D slot. Slot r, lane l -> row r0 + r + 8*lhalf,
    // column jbase + l16 (column depends only on the lane).
    float best[8];
    int   bestj[8];
#pragma unroll
    for (int r = 0; r < 8; ++r) { best[r] = FLT_MAX; bestj[r] = 0; }

    for (int base = 0; base < Ny; base += CHUNK) {
        __syncthreads();
        // Cooperative stage of (-2*y, |y|^2) into LDS.
        for (int t = tid; t < CHUNK; t += WG_THREADS) {
            const int j = base + t;
            const float y0 = Yb[(size_t)j * 3 + 0];
            const float y1 = Yb[(size_t)j * 3 + 1];
            const float y2 = Yb[(size_t)j * 3 + 2];
            sY[t] = make_float4(-2.0f * y0, -2.0f * y1, -2.0f * y2,
                                y0 * y0 + y1 * y1 + y2 * y2);
        }
        __syncthreads();

#pragma unroll 8
        for (int jt = 0; jt < TILES_PER_CHUNK; ++jt) {
            const float4 sy = sY[jt * 16 + l16];     // 128-bit ds_load, conflict-free

            // B matrix (4x16 fp32): row K striped across 16 lanes per half-VGPR:
            //   vgpr0: K0 (lanes 0-15) / K2 (lanes 16-31)
            //   vgpr1: K1 (lanes 0-15) / K3 (lanes 16-31)
            v2f bm;
            bm.x = lhalf ? sy.z : sy.x;              // -2y2  : -2y0
            bm.y = lhalf ? sy.w : sy.y;              // |y|^2 : -2y1

            v8f c = {};                              // SRC2 = inline 0
            c = __builtin_amdgcn_wmma_f32_16x16x4_f32(
                    /*neg_a=*/false, a, /*neg_b=*/false, bm,
                    /*c_mod=*/(short)0, c, /*reuse_a=*/false, /*reuse_b=*/false);

            const int jglob = base + jt * 16 + l16;
#pragma unroll
            for (int r = 0; r < 8; ++r) {
                const float dv = c[r];
                const bool lt = dv < best[r];
                best[r]  = lt ? dv    : best[r];     // v_cndmask: EXEC stays all-1s
                bestj[r] = lt ? jglob : bestj[r];
            }
        }
    }

    // Row-min: butterfly over the 16 lanes of each half (masks 1,2,4,8 stay
    // inside the half). Tie-break toward the smaller column index (matches argmin).
#pragma unroll
    for (int m = 1; m < 16; m <<= 1) {
#pragma unroll
        for (int r = 0; r < 8; ++r) {
            const float ov = __shfl_xor(best[r], m, 32);
            const int   oj = __shfl_xor(bestj[r], m, 32);
            const bool take = (ov < best[r]) || (ov == best[r] && oj < bestj[r]);
            best[r]  = take ? ov : best[r];
            bestj[r] = take ? oj : bestj[r];
        }
    }

    // Re-add the per-row |x|^2 shift (held by lane 8*lhalf + r) and clamp.
    if (l16 == 0) {
        const int rowbase = r0 + 8 * lhalf;
#pragma unroll
        for (int r = 0; r < 8; ++r) {
            const float xnr = __shfl(xn, 8 * lhalf + r, 32);
            dist[(size_t)b * Nx + rowbase + r] = fmaxf(best[r] + xnr, 0.0f);
            idx [(size_t)b * Nx + rowbase + r] = bestj[r];
        }
    }
}

extern "C" void kernel_launch(void* const* d_in, const int* in_sizes, int n_in,
                              void* d_out, int out_size, void* d_ws, size_t ws_size,
                              hipStream_t stream) {
    (void)n_in; (void)d_ws; (void)ws_size; (void)out_size;

    const float* xyz1 = (const float*)d_in[0];
    const float* xyz2 = (const float*)d_in[1];

    const int N = 4096, M = 4096;
    const int B = in_sizes[0] / (N * 3);

    float* out_f = (float*)d_out;
    int*   out_i = (int*)d_out;
    float* dist1 = out_f;                       // [B,N]
    float* dist2 = out_f + (size_t)B * N;       // [B,M]
    int*   idx1  = out_i + (size_t)2 * B * N;   // [B,N]
    int*   idx2  = out_i + (size_t)2 * B * N + (size_t)B * M;  // [B,M]

    dim3 block(WG_THREADS, 1, 1);
    dim3 grid1(N / (16 * WAVES_PER_WG), B, 1);
    dim3 grid2(M / (16 * WAVES_PER_WG), B, 1);

    // Pass 1: for each x in xyz1, NN in xyz2.
    chamfer_nn_dir_kernel<<<grid1, block, 0, stream>>>(xyz1, xyz2, dist1, idx1, N, M);
    // Pass 2: roles swapped.
    chamfer_nn_dir_kernel<<<grid2, block, 0, stream>>>(xyz2, xyz1, dist2, idx2, M, N);
}